// LINK_28991029248697
// MI455X (gfx1250) — compile-verified
//
#include <hip/hip_runtime.h>
#include <math.h>
#include <stdint.h>

#define C_CLASSES 64
#define WAVES_PER_BLOCK 8   // 256 threads / wave32
#define ASYNC_DEPTH 4       // async gather pipeline depth (power of two)

// ---- gfx1250 async global->LDS path (ASYNCcnt-tracked), guarded -----------
#if defined(__has_builtin)
# if __has_builtin(__builtin_amdgcn_global_load_async_to_lds_b64) && \
     __has_builtin(__builtin_amdgcn_s_wait_asynccnt)
#  define GCN_HAVE_ASYNC_LDS 1
# endif
#endif
#ifndef GCN_HAVE_ASYNC_LDS
# define GCN_HAVE_ASYNC_LDS 0
#endif

#if GCN_HAVE_ASYNC_LDS
// Param type per hipcc diagnostic: pointer to 8-byte int vector
typedef int v2i_t __attribute__((vector_size(2 * sizeof(int))));
typedef __attribute__((address_space(1))) v2i_t* g_v2i_p;
typedef __attribute__((address_space(3))) v2i_t* l_v2i_p;

__device__ __forceinline__ void async_copy_f2(const float* g, float2* l) {
  // global_load_async_to_lds_b64: per-lane 8B global -> per-lane LDS slot
  __builtin_amdgcn_global_load_async_to_lds_b64(
      (g_v2i_p)(g), (l_v2i_p)(l), 0 /*imm offset*/, 0 /*cpol*/);
}
#endif

// ---- wave32 reductions -----------------------------------------------------
__device__ __forceinline__ float wave_max(float v) {
#pragma unroll
  for (int s = 16; s >= 1; s >>= 1) v = fmaxf(v, __shfl_xor(v, s, 32));
  return v;
}
__device__ __forceinline__ float wave_sum(float v) {
#pragma unroll
  for (int s = 16; s >= 1; s >>= 1) v += __shfl_xor(v, s, 32);
  return v;
}

// ---- CSR build -------------------------------------------------------------
__global__ void zero_i32(int* __restrict__ p, int n) {
  int i = blockIdx.x * blockDim.x + threadIdx.x;
  if (i < n) p[i] = 0;
}

__global__ void count_rows(const int* __restrict__ rows, int* __restrict__ cnt, int nE) {
  int stride = gridDim.x * blockDim.x;
  for (int e = blockIdx.x * blockDim.x + threadIdx.x; e < nE; e += stride)
    atomicAdd(&cnt[rows[e]], 1);
}

// single-workgroup chunked exclusive scan over N (~100K ints: negligible time)
__global__ __launch_bounds__(1024) void scan_offsets(const int* __restrict__ cnt,
                                                     int* __restrict__ off,
                                                     int* __restrict__ cur, int n) {
  __shared__ int sm[1024];
  __shared__ int sbase;
  const int tid = threadIdx.x;
  if (tid == 0) sbase = 0;
  __syncthreads();
  for (int base = 0; base < n; base += 1024) {
    const int i = base + tid;
    const int v = (i < n) ? cnt[i] : 0;
    sm[tid] = v;
    __syncthreads();
#pragma unroll
    for (int d = 1; d < 1024; d <<= 1) {      // Hillis-Steele inclusive scan
      int t = (tid >= d) ? sm[tid - d] : 0;
      __syncthreads();
      sm[tid] += t;
      __syncthreads();
    }
    const int excl = sbase + sm[tid] - v;
    if (i < n) { off[i] = excl; cur[i] = excl; }
    __syncthreads();
    if (tid == 1023) sbase += sm[1023];
    __syncthreads();
  }
}

__global__ void fill_csr(const int* __restrict__ rows, const int* __restrict__ cols,
                         int* __restrict__ cur, int* __restrict__ csr_col, int nE) {
  int stride = gridDim.x * blockDim.x;
  for (int e = blockIdx.x * blockDim.x + threadIdx.x; e < nE; e += stride) {
    int pos = atomicAdd(&cur[rows[e]], 1);
    csr_col[pos] = cols[e];
  }
}

// ---- fused aggregate (atomic-free) + log_softmax: one wave per node --------
// Gather of neighbor rows of Wt is pipelined through LDS with
// global_load_async_to_lds_b64 at depth ASYNC_DEPTH (ASYNCcnt discipline).
__global__ __launch_bounds__(256) void aggregate_logsoftmax(
    const float* __restrict__ Wt, const float* __restrict__ bias,
    const int* __restrict__ off, const int* __restrict__ cnt,
    const int* __restrict__ csr_col, float* __restrict__ out, int nN) {
#if GCN_HAVE_ASYNC_LDS
  __shared__ float2 stage[WAVES_PER_BLOCK][ASYNC_DEPTH][32];  // 8KB circular buffers
#endif
  const int lane = threadIdx.x & 31;
  const int wave = threadIdx.x >> 5;
  const int node = blockIdx.x * WAVES_PER_BLOCK + wave;
  if (node >= nN) return;                 // wave-uniform exit

  const int start = off[node];
  const int deg   = cnt[node];
  const int c0    = lane << 1;            // each lane owns classes c0, c0+1
  float ax = 0.f, ay = 0.f;

#if GCN_HAVE_ASYNC_LDS
  // Prologue: fill the pipeline.
  const int pre = (deg < ASYNC_DEPTH) ? deg : ASYNC_DEPTH;
  for (int k = 0; k < pre; ++k) {
    const int col = csr_col[start + k];
    async_copy_f2(Wt + (size_t)col * C_CLASSES + c0, &stage[wave][k][lane]);
  }
  int k = 0;
  // Steady state: wait(DEPTH-1) => in-order completion guarantees row k landed
  // (if row k were still in flight, all DEPTH rows would be => cnt == DEPTH).
  for (; k + ASYNC_DEPTH < deg; ++k) {
    __builtin_amdgcn_s_wait_asynccnt(ASYNC_DEPTH - 1);
    const float2 v = stage[wave][k & (ASYNC_DEPTH - 1)][lane];
    ax += v.x; ay += v.y;                 // buffer k is now free
    const int col = csr_col[start + k + ASYNC_DEPTH];
    async_copy_f2(Wt + (size_t)col * C_CLASSES + c0,
                  &stage[wave][(k + ASYNC_DEPTH) & (ASYNC_DEPTH - 1)][lane]);
  }
  // Tail: <= DEPTH rows buffered; one wait drains them all.
  if (k < deg) {
    __builtin_amdgcn_s_wait_asynccnt(0);
    for (; k < deg; ++k) {
      const float2 v = stage[wave][k & (ASYNC_DEPTH - 1)][lane];
      ax += v.x; ay += v.y;
    }
  }
#else
  int k = 0;
  for (; k + 4 <= deg; k += 4) {          // unroll-4: 4 in-flight b64 loads
    const int i0 = csr_col[start + k + 0];
    const int i1 = csr_col[start + k + 1];
    const int i2 = csr_col[start + k + 2];
    const int i3 = csr_col[start + k + 3];
    const float2 v0 = *(const float2*)(Wt + (size_t)i0 * C_CLASSES + c0);
    const float2 v1 = *(const float2*)(Wt + (size_t)i1 * C_CLASSES + c0);
    const float2 v2 = *(const float2*)(Wt + (size_t)i2 * C_CLASSES + c0);
    const float2 v3 = *(const float2*)(Wt + (size_t)i3 * C_CLASSES + c0);
    ax += (v0.x + v1.x) + (v2.x + v3.x);
    ay += (v0.y + v1.y) + (v2.y + v3.y);
  }
  for (; k < deg; ++k) {
    const int i0 = csr_col[start + k];
    const float2 v0 = *(const float2*)(Wt + (size_t)i0 * C_CLASSES + c0);
    ax += v0.x; ay += v0.y;
  }
#endif

  ax += bias[c0];
  ay += bias[c0 + 1];
  const float m   = wave_max(fmaxf(ax, ay));
  const float s   = wave_sum(expf(ax - m) + expf(ay - m));
  const float lse = m + logf(s);
  float2 r; r.x = ax - lse; r.y = ay - lse;
  *(float2*)(out + (size_t)node * C_CLASSES + c0) = r;
}

// ---- fallback path (scratch too small): plain f32 atomic scatter -----------
__global__ void zero_f32(float* __restrict__ p, size_t n) {
  size_t stride = (size_t)gridDim.x * blockDim.x;
  for (size_t i = (size_t)blockIdx.x * blockDim.x + threadIdx.x; i < n; i += stride)
    p[i] = 0.f;
}

__global__ void scatter_edges_atomic(const int* __restrict__ rows, const int* __restrict__ cols,
                                     const float* __restrict__ Wt, float* __restrict__ out, int nE) {
  const size_t total  = (size_t)nE * 32;         // 32 threads x float2 per edge
  const size_t stride = (size_t)gridDim.x * blockDim.x;
  for (size_t i = (size_t)blockIdx.x * blockDim.x + threadIdx.x; i < total; i += stride) {
    const int e  = (int)(i >> 5);
    const int c0 = (int)(i & 31) << 1;
    const int r  = rows[e];
    const int c  = cols[e];
    const float2 v = *(const float2*)(Wt + (size_t)c * C_CLASSES + c0);
    atomicAdd(out + (size_t)r * C_CLASSES + c0,     v.x);
    atomicAdd(out + (size_t)r * C_CLASSES + c0 + 1, v.y);
  }
}

__global__ __launch_bounds__(256) void logsoftmax_inplace(const float* __restrict__ bias,
                                                          float* __restrict__ out, int nN) {
  const int lane = threadIdx.x & 31;
  const int wave = threadIdx.x >> 5;
  const int node = blockIdx.x * WAVES_PER_BLOCK + wave;
  if (node >= nN) return;
  const int c0 = lane << 1;
  const float2 v = *(const float2*)(out + (size_t)node * C_CLASSES + c0);
  const float ax = v.x + bias[c0];
  const float ay = v.y + bias[c0 + 1];
  const float m   = wave_max(fmaxf(ax, ay));
  const float s   = wave_sum(expf(ax - m) + expf(ay - m));
  const float lse = m + logf(s);
  float2 r; r.x = ax - lse; r.y = ay - lse;
  *(float2*)(out + (size_t)node * C_CLASSES + c0) = r;
}

// ---- host launcher ---------------------------------------------------------
extern "C" void kernel_launch(void* const* d_in, const int* in_sizes, int n_in,
                              void* d_out, int out_size, void* d_ws, size_t ws_size,
                              hipStream_t stream) {
  (void)n_in; (void)out_size;
  const int*   edges = (const int*)d_in[0];     // [2, E] int32 (JAX x64-off)
  const float* Wt    = (const float*)d_in[1];   // [N, C] f32
  const float* bias  = (const float*)d_in[2];   // [C]    f32
  const int nE = in_sizes[0] / 2;
  const int nN = in_sizes[1] / C_CLASSES;
  const int* rows = edges;
  const int* cols = edges + nE;
  float* out = (float*)d_out;

  const size_t need = (size_t)nN * 3 * sizeof(int) + (size_t)nE * sizeof(int);
  const int edge_blocks = ((nE + 255) / 256) < 2048 ? ((nE + 255) / 256) : 2048;

  if (d_ws && ws_size >= need) {
    int* cnt = (int*)d_ws;      // [N] degrees
    int* off = cnt + nN;        // [N] CSR row starts
    int* cur = off + nN;        // [N] fill cursors
    int* csr = cur + nN;        // [E] CSR column indices

    zero_i32<<<(nN + 255) / 256, 256, 0, stream>>>(cnt, nN);
    count_rows<<<edge_blocks, 256, 0, stream>>>(rows, cnt, nE);
    scan_offsets<<<1, 1024, 0, stream>>>(cnt, off, cur, nN);
    fill_csr<<<edge_blocks, 256, 0, stream>>>(rows, cols, cur, csr, nE);
    aggregate_logsoftmax<<<(nN + WAVES_PER_BLOCK - 1) / WAVES_PER_BLOCK, 256, 0, stream>>>(
        Wt, bias, off, cnt, csr, out, nN);
  } else {
    zero_f32<<<2048, 256, 0, stream>>>(out, (size_t)nN * C_CLASSES);
    scatter_edges_atomic<<<4096, 256, 0, stream>>>(rows, cols, Wt, out, nE);
    logsoftmax_inplace<<<(nN + WAVES_PER_BLOCK - 1) / WAVES_PER_BLOCK, 256, 0, stream>>>(
        bias, out, nN);
  }
}